// ExpectedSignature_12841952215247
// MI455X (gfx1250) — compile-verified
//
#include <hip/hip_runtime.h>
#include <hip/hip_bf16.h>

// ---- problem constants (match reference) ----
#define M_LVL   6
#define SIGLEN  19530
#define BB      64
#define SS      32
#define ROWS    (BB * SS)          // 2048
#define NBISECT 100
#define C_PARAM 4.0f
#define A_PARAM 1.0f

#define CHUNK   1024                                   // K-coords per wave chunk
#define NCHUNK  ((SIGLEN + CHUNK - 1) / CHUNK)         // 20
#define NGROUP  (ROWS / 16)                            // 128 row-groups of 16

typedef __attribute__((ext_vector_type(2))) float v2f;
typedef __attribute__((ext_vector_type(8))) float v8f;

__device__ __forceinline__ int lvl_of(int i) {
    // level id 1..6 for signature coordinate i (cumulative starts 0,5,30,155,780,3905)
    return 1 + (i >= 5) + (i >= 30) + (i >= 155) + (i >= 780) + (i >= 3905);
}

__device__ __forceinline__ bool finitef(float v) {
    return (v == v) && (fabsf(v) <= 3.402823466e38f);
}

__device__ __forceinline__ v8f wmma_acc(v2f a, v2f b, v8f c) {
    return __builtin_amdgcn_wmma_f32_16x16x4_f32(false, a, false, b, (short)0, c, false, false);
}

// ---------------------------------------------------------------------------
// Kernel 1: per-level squared-norm sums via V_WMMA_F32_16X16X4_F32.
// Levels are contiguous coordinate ranges, so per (chunk x level) segment the
// one-hot B fragment is loop-invariant. Interior granules run a branch-free,
// depth-2 software-pipelined loop: load_b64 -> pk_mul -> wmma. Segment-edge
// granules (<=2 per segment) take a masked slow path.
// Grid: (NCHUNK, NGROUP/4), block (32,4): one wave per (chunk, group).
// ---------------------------------------------------------------------------
__global__ __launch_bounds__(128)
void level_sums_wmma(const float* __restrict__ x, float* __restrict__ partials) {
    const int lane  = threadIdx.x;                       // 0..31
    const int group = blockIdx.y * 4 + threadIdx.y;      // 0..127
    const int chunk = blockIdx.x;                        // 0..19

    const int rowM = lane & 15;                          // M index 0..15
    const int koff = (lane >> 4) * 2;                    // K: lanes 0-15 -> {0,1}, 16-31 -> {2,3}
    const int col  = lane & 15;                          // N (level-1) column for B/C
    const float* rowPtr = x + (long)(group * 16 + rowM) * SIGLEN;

    v8f acc = {0.f, 0.f, 0.f, 0.f, 0.f, 0.f, 0.f, 0.f};

    const int cStart = chunk * CHUNK;
    const int cEnd   = min(cStart + CHUNK, SIGLEN);
    const int lvlStart[M_LVL + 1] = {0, 5, 30, 155, 780, 3905, SIGLEN};

#pragma unroll
    for (int k = 1; k <= M_LVL; ++k) {
        const int segBeg = max(cStart, lvlStart[k - 1]);
        const int segEnd = min(cEnd,   lvlStart[k]);
        if (segBeg >= segEnd) continue;

        // Loop-invariant one-hot B fragment for this level.
        v2f bfrag;
        const float bval = (col == k - 1) ? 1.0f : 0.0f;
        bfrag[0] = bval; bfrag[1] = bval;

        const int g0       = segBeg & ~3;                // granule holding segBeg
        const int g1       = (segEnd + 3) & ~3;          // exclusive granule end
        const int gFastBeg = (segBeg + 3) & ~3;          // first fully-interior granule
        const int gFastEnd = segEnd & ~3;                // end of fully-interior granules

        // --- head partial granule (rare) ---
        for (int g = g0; g < min(gFastBeg, g1); g += 4) {
            const int co = g + koff;
            float ax = 0.f, ay = 0.f;
            if (co < SIGLEN) {
                const float2 v = *(const float2*)(rowPtr + co);
                ax = (co     >= segBeg && co     < segEnd) ? v.x * v.x : 0.f;
                ay = (co + 1 >= segBeg && co + 1 < segEnd) ? v.y * v.y : 0.f;
            }
            v2f a; a[0] = ax; a[1] = ay;
            acc = wmma_acc(a, bfrag, acc);
        }

        // --- interior: branch-free, depth-2 pipelined ---
        if (gFastBeg < gFastEnd) {
            const float* p = rowPtr + gFastBeg + koff;
            float2 vcur = *(const float2*)p;
            for (int g = gFastBeg; g + 4 < gFastEnd; g += 4) {
                p += 4;
                const float2 vnxt = *(const float2*)p;   // issue next load before WMMA
                v2f a; a[0] = vcur.x * vcur.x; a[1] = vcur.y * vcur.y;
                acc = wmma_acc(a, bfrag, acc);
                vcur = vnxt;
            }
            v2f a; a[0] = vcur.x * vcur.x; a[1] = vcur.y * vcur.y;
            acc = wmma_acc(a, bfrag, acc);
        }

        // --- tail partial granule (rare) ---
        for (int g = max(gFastBeg, gFastEnd); g < g1; g += 4) {
            const int co = g + koff;
            float ax = 0.f, ay = 0.f;
            if (co < SIGLEN) {
                const float2 v = *(const float2*)(rowPtr + co);
                ax = (co     >= segBeg && co     < segEnd) ? v.x * v.x : 0.f;
                ay = (co + 1 >= segBeg && co + 1 < segEnd) ? v.y * v.y : 0.f;
            }
            v2f a; a[0] = ax; a[1] = ay;
            acc = wmma_acc(a, bfrag, acc);
        }
    }

    // C layout: VGPR v, lanes 0-15 -> M=v, lanes 16-31 -> M=v+8; N = lane&15.
    if (col < M_LVL) {
        const int mhalf = (lane >> 4) * 8;
#pragma unroll
        for (int v = 0; v < 8; ++v) {
            const int row = group * 16 + mhalf + v;
            partials[((long)chunk * ROWS + row) * 8 + col] = acc[v];
        }
    }
}

// ---------------------------------------------------------------------------
// Kernel 2: reduce chunk partials, phi, 100-step bisection, write root^k table.
// One thread per (b,s) row.
// ---------------------------------------------------------------------------
__global__ __launch_bounds__(256)
void bisect_kernel(const float* __restrict__ partials, float* __restrict__ rootpow) {
    const int row = blockIdx.x * blockDim.x + threadIdx.x;
    if (row >= ROWS) return;

    float ck[M_LVL];
#pragma unroll
    for (int k = 0; k < M_LVL; ++k) ck[k] = 0.f;
    for (int ch = 0; ch < NCHUNK; ++ch) {
#pragma unroll
        for (int k = 0; k < M_LVL; ++k)
            ck[k] += partials[((long)ch * ROWS + row) * 8 + k];
    }

    float nq = 1.f;
#pragma unroll
    for (int k = 0; k < M_LVL; ++k) nq += ck[k];

    float phi = nq;
    if (nq > C_PARAM) {
        phi = C_PARAM + powf(C_PARAM, 1.f + A_PARAM) *
                        (powf(C_PARAM, -A_PARAM) - powf(nq, -A_PARAM)) / A_PARAM;
    }
    const float c0 = 1.f - phi;

    bool finite = finitef(c0);
#pragma unroll
    for (int k = 0; k < M_LVL; ++k) finite = finite && finitef(ck[k]);

    float lo = 0.f, hi = 2.f;
    for (int it = 0; it < NBISECT; ++it) {
        const float mid = 0.5f * (lo + hi);
        const float t2 = mid * mid;
        float p = c0, tp = 1.f;
#pragma unroll
        for (int k = 0; k < M_LVL; ++k) { tp *= t2; p += ck[k] * tp; }
        const bool neg = p < 0.f;
        lo = neg ? mid : lo;
        hi = neg ? hi : mid;
    }
    float root = 0.5f * (lo + hi);
    if (!finite) root = 0.f;
    root = fminf(root, 1.f);

    float rp = 1.f;
    rootpow[row * 8 + 0] = 1.f;
#pragma unroll
    for (int k = 1; k <= M_LVL; ++k) { rp *= root; rootpow[row * 8 + k] = rp; }
    rootpow[row * 8 + 7] = 0.f;
}

// ---------------------------------------------------------------------------
// Kernel 3: out[b,l] = (1/S) * sum_s x[b,s,l] * root[b,s]^level(l)
// One block per (l-slab, b); rootpow staged in LDS. float2 traffic (rows are
// only 8-byte aligned: 19530*4 % 16 == 8). Second read of x should largely
// hit the 192 MB L2 (input is 160 MB).
// ---------------------------------------------------------------------------
__global__ __launch_bounds__(256)
void scale_mean_kernel(const float* __restrict__ x, const float* __restrict__ rootpow,
                       float* __restrict__ out) {
    __shared__ float rp[SS * 8];
    const int b   = blockIdx.y;
    const int tid = threadIdx.x;
    rp[tid] = rootpow[(long)b * SS * 8 + tid];           // 256 == SS*8 exactly
    __syncthreads();

    const int p = blockIdx.x * 256 + tid;                // float2 index
    const int halfL = SIGLEN / 2;                        // 9765
    if (p >= halfL) return;
    const int l0 = p * 2;
    const int k0 = lvl_of(l0);
    const int k1 = lvl_of(l0 + 1);

    const float* xb = x + (long)b * SS * SIGLEN + l0;
    float accx = 0.f, accy = 0.f;
#pragma unroll 4
    for (int s = 0; s < SS; ++s) {
        const float2 v = *(const float2*)(xb + (long)s * SIGLEN);
        accx += v.x * rp[s * 8 + k0];
        accy += v.y * rp[s * 8 + k1];
    }
    out[(long)b * SIGLEN + l0]     = accx * (1.f / SS);
    out[(long)b * SIGLEN + l0 + 1] = accy * (1.f / SS);
}

extern "C" void kernel_launch(void* const* d_in, const int* in_sizes, int n_in,
                              void* d_out, int out_size, void* d_ws, size_t ws_size,
                              hipStream_t stream) {
    const float* x = (const float*)d_in[0];
    float* out = (float*)d_out;

    float* partials = (float*)d_ws;                         // NCHUNK*ROWS*8 floats (~1.25 MB)
    float* rootpow  = partials + (long)NCHUNK * ROWS * 8;   // ROWS*8 floats (64 KB)

    {   // per-level sums (WMMA)
        dim3 blk(32, 4), grd(NCHUNK, NGROUP / 4);
        level_sums_wmma<<<grd, blk, 0, stream>>>(x, partials);
    }
    {   // reduce + bisection roots
        bisect_kernel<<<ROWS / 256, 256, 0, stream>>>(partials, rootpow);
    }
    {   // scale by root^level and mean over samples
        dim3 blk(256), grd((SIGLEN / 2 + 255) / 256, BB);
        scale_mean_kernel<<<grd, blk, 0, stream>>>(x, rootpow, out);
    }
}